// TorchTEFScorer_80994493268395
// MI455X (gfx1250) — compile-verified
//
#include <hip/hip_runtime.h>
#include <math.h>

typedef __attribute__((ext_vector_type(2))) float v2f;
typedef __attribute__((ext_vector_type(8))) float v8f;

#define B_DIM 8
#define S_DIM 8192
#define H_DIM 1024
#define BS_DIM (B_DIM * S_DIM)
#define THRESH 0.95f

// ---------------------------------------------------------------------------
// Kernel 1: logits = hs @ w + b, gates = sigmoid(logits)
// One wave32 per 16-row tile; K-loop of V_WMMA_F32_16X16X4_F32 (full fp32).
// B-matrix (w chunk) is replicated across all 16 N columns, so every column
// of D equals the dot product -> exact result, B column layout irrelevant.
// ---------------------------------------------------------------------------
__global__ __launch_bounds__(256) void matvec_wmma_kernel(
    const float* __restrict__ hs,
    const float* __restrict__ w,
    const float* __restrict__ bias_p,
    float* __restrict__ logits,
    float* __restrict__ gates)
{
    __shared__ float wlds[H_DIM];  // 4 KB broadcast copy of w
    for (int i = threadIdx.x; i < H_DIM; i += 256) wlds[i] = w[i];
    __syncthreads();

    const int wave = threadIdx.x >> 5;
    const int lane = threadIdx.x & 31;
    const int tile = blockIdx.x * 8 + wave;   // 16 rows per wave
    const int m    = lane & 15;               // row within tile (A layout)
    const int h    = lane >> 4;               // half-wave selects K pair

    const float* rowp = hs + (size_t)(tile * 16 + m) * H_DIM;

    // Two accumulators to break the WMMA->WMMA accumulation RAW chain.
    v8f c0 = {};
    v8f c1 = {};
    for (int k0 = 0; k0 < H_DIM; k0 += 8) {
        // A 16x4 fp32 layout: lane(m,h), VGPR v holds K = v + 2h.
        v2f a0 = *(const v2f*)(rowp + k0 + 2 * h);
        v2f b0 = *(const v2f*)(wlds + k0 + 2 * h);
        c0 = __builtin_amdgcn_wmma_f32_16x16x4_f32(
                 false, a0, false, b0, (short)0, c0, false, false);
        v2f a1 = *(const v2f*)(rowp + k0 + 4 + 2 * h);
        v2f b1 = *(const v2f*)(wlds + k0 + 4 + 2 * h);
        c1 = __builtin_amdgcn_wmma_f32_16x16x4_f32(
                 false, a1, false, b1, (short)0, c1, false, false);
    }
    v8f c = c0 + c1;

    // D layout: lane 0 (N=0) holds M=0..7 in c[0..7]; lane 16 holds M=8..15.
    if ((lane & 15) == 0) {
        const float bb   = bias_p[0];
        const int   base = tile * 16 + h * 8;
#pragma unroll
        for (int i = 0; i < 8; ++i) {
            float lg = c[i] + bb;
            logits[base + i] = lg;
            gates[base + i]  = 1.0f / (1.0f + __expf(-lg));
        }
    }
}

// ---------------------------------------------------------------------------
// Kernel 2: cumulative-threshold keep mask, sort-free formulation.
// 32 blocks per batch row; each block stages the full 8192-wide gated row in
// LDS (32 KB), reduces the row total in-block, then each thread evaluates
// one element j against all i via broadcast LDS reads.
// ---------------------------------------------------------------------------
__global__ __launch_bounds__(256) void keep_mask_kernel(
    const float* __restrict__ gates,
    const int* __restrict__ mask,
    float* __restrict__ keep)
{
    __shared__ __align__(16) float g[S_DIM];   // gated row (32 KB)
    __shared__ float red[256];

    const int    batch = blockIdx.x >> 5;
    const int    seg   = blockIdx.x & 31;
    const int    tid   = threadIdx.x;
    const size_t rb    = (size_t)batch * S_DIM;

    float part = 0.0f;
    for (int i = tid; i < S_DIM; i += 256) {
        float gg = mask[rb + i] ? gates[rb + i] : 0.0f;
        g[i] = gg;
        part += gg;
    }
    red[tid] = part;
    __syncthreads();
    for (int off = 128; off > 0; off >>= 1) {
        if (tid < off) red[tid] += red[tid + off];
        __syncthreads();
    }
    const float total = fmaxf(red[0], 1e-12f);   // max(sum, EPS)
    const float limit = THRESH * total;          // compare in gated space

    const int   j    = seg * 256 + tid;
    const float gj   = g[j];
    const int   actj = mask[rb + j];

    // cumExcl = sum of elements strictly ahead of j in stable descending sort
    float cumExcl = 0.0f;
    const float4* g4 = (const float4*)g;
#pragma unroll 4
    for (int i4 = 0; i4 < S_DIM / 4; ++i4) {
        float4 v  = g4[i4];
        int    ib = i4 * 4;
        cumExcl += ((v.x > gj) || ((v.x == gj) && (ib + 0 < j))) ? v.x : 0.0f;
        cumExcl += ((v.y > gj) || ((v.y == gj) && (ib + 1 < j))) ? v.y : 0.0f;
        cumExcl += ((v.z > gj) || ((v.z == gj) && (ib + 2 < j))) ? v.z : 0.0f;
        cumExcl += ((v.w > gj) || ((v.w == gj) && (ib + 3 < j))) ? v.w : 0.0f;
    }

    // cumExcl == 0  <=>  j is sorted position 0 (always kept)
    bool kept = actj && (((cumExcl + gj) <= limit) || (cumExcl == 0.0f));
    keep[rb + j] = kept ? 1.0f : 0.0f;
}

// ---------------------------------------------------------------------------
extern "C" void kernel_launch(void* const* d_in, const int* in_sizes, int n_in,
                              void* d_out, int out_size, void* d_ws, size_t ws_size,
                              hipStream_t stream) {
    (void)in_sizes; (void)n_in; (void)out_size; (void)d_ws; (void)ws_size;

    const float* hs   = (const float*)d_in[0];  // [8, 8192, 1024] f32
    const int*   mask = (const int*)d_in[1];    // [8, 8192] i32
    const float* w    = (const float*)d_in[2];  // [1024] f32
    const float* b    = (const float*)d_in[3];  // [1] f32

    float* out    = (float*)d_out;              // [logits | gates | keep]
    float* logits = out;
    float* gates  = out + BS_DIM;
    float* keep   = out + 2 * BS_DIM;

    // 65536 rows / 16 rows-per-wave / 8 waves-per-block = 512 blocks
    matvec_wmma_kernel<<<BS_DIM / 16 / 8, 256, 0, stream>>>(hs, w, b, logits, gates);

    // 8 batches * 32 segments = 256 blocks of 256 threads (one j each)
    keep_mask_kernel<<<B_DIM * 32, 256, 0, stream>>>(gates, mask, keep);
}